// MergingModel_45672682226372
// MI455X (gfx1250) — compile-verified
//
#include <hip/hip_runtime.h>
#include <math.h>

#define N_IMG   8
#define R_REFL  16384
#define H_HARM  2
#define OPS_N   2
#define MC      32
#define NREFL   262144
#define DHID    128
#define NR      (N_IMG * R_REFL)        /* 131072 */
#define NGRP    (OPS_N * NR)            /* 262144 */
#define NOBS    (NGRP * H_HARM)         /* 524288 */
#define LOG2PI  1.8378770664093453f

typedef __attribute__((ext_vector_type(2))) float v2f;
typedef __attribute__((ext_vector_type(8))) float v8f;

static __device__ inline v8f wmma4(v2f a, v2f b, v8f c) {
    // D = A(16x4,f32) x B(4x16,f32) + C(16x16,f32)
    return __builtin_amdgcn_wmma_f32_16x16x4_f32(false, a, false, b, (short)0, c, false, false);
}

// ---------------- Kernel 1: z = loc + softplus(raw_scale)*eps, Imodel, KL partials
__global__ __launch_bounds__(256) void k_sample(const float* __restrict__ loc,
                                                const float* __restrict__ raw_scale,
                                                const float* __restrict__ eps,
                                                float* __restrict__ zT,
                                                float* __restrict__ Imodel,
                                                float* __restrict__ klp) {
    __shared__ float sm[256];
    const int i = blockIdx.x * 256 + threadIdx.x;
    const float lc = loc[i];
    const float x  = raw_scale[i];
    const float qs = (x > 20.f) ? x : log1pf(expf(x));
    Imodel[2 * i]     = lc;
    Imodel[2 * i + 1] = qs;
#pragma unroll 8
    for (int m = 0; m < MC; ++m)
        zT[(size_t)i * MC + m] = lc + qs * eps[(size_t)m * NREFL + i];
    float kv = 0.5f * (qs * qs + lc * lc - 1.f) - logf(qs);
    sm[threadIdx.x] = kv; __syncthreads();
    for (int s = 128; s > 0; s >>= 1) { if (threadIdx.x < s) sm[threadIdx.x] += sm[threadIdx.x + s]; __syncthreads(); }
    if (threadIdx.x == 0) klp[blockIdx.x] = sm[0];
}

// ---------------- Kernel 2: fused feats -> GEMM1(relu) -> GEMM2 -> z-gather -> h-sum -> loglik
__global__ __launch_bounds__(128) void k_main(
    const float* __restrict__ I, const float* __restrict__ SigI,
    const float* __restrict__ metadata, const int* __restrict__ refl_id,
    const float* __restrict__ wav_exp,
    const float* __restrict__ W1, const float* __restrict__ b1,
    const float* __restrict__ W2, const float* __restrict__ b2,
    const float* __restrict__ zT, const float* __restrict__ Imodel,
    float* __restrict__ ws_ll, float* __restrict__ ws_ipm, float* __restrict__ ws_maskf)
{
    // K-pair-packed weights: element p,col = (W[2p][col], W[2p+1][col]) -> one ds_load_b64 per B frag
    __shared__ v2f  W1pS[8 * DHID];           // 8 KB  (K padded 14->16)
    __shared__ v2f  W2pS[64 * MC];            // 16 KB
    __shared__ float hidS[4][16][130];        // 32.5 KB, stride 130: bank-spread, rows 8B-aligned
    __shared__ union {                        // feats dead after GEMM1; ipred born in GEMM2
        float feats[4][16][20];               // 5 KB, stride 20: bank-spread, rows 8B-aligned
        float ipred[4][8][32];                // 4 KB
    } __align__(16) fip;
    __shared__ int  ridS[4][16];
    __shared__ int  mskS[4][16];

    const int t = threadIdx.x;
    // Stage W1 as K-pairs (pad K 14->16 with zeros)
    for (int e = t; e < 8 * DHID; e += 128) {
        const int p = e >> 7, col = e & 127;
        v2f v;
        v.x = (2 * p     < 14) ? W1[(2 * p)     * DHID + col] : 0.f;
        v.y = (2 * p + 1 < 14) ? W1[(2 * p + 1) * DHID + col] : 0.f;
        W1pS[e] = v;
    }
    // Stage W2 as K-pairs
    for (int e = t; e < 64 * MC; e += 128) {
        const int p = e >> 5, col = e & 31;
        v2f v;
        v.x = W2[(2 * p)     * MC + col];
        v.y = W2[(2 * p + 1) * MC + col];
        W2pS[e] = v;
    }

    const int w = t >> 5, l = t & 31;
    const int half = l >> 4, li = l & 15;
    const int base = blockIdx.x * 64 + w * 16;

    // Build features: each row by two lanes (low half: f0..f7, high half: f8..f15)
    {
        const int m   = li;
        const int o   = base + m;
        const int grp = o >> 1;
        const int nr  = grp & (NR - 1);
        if (half == 0) {
            int rr  = refl_id[o];
            int mk  = (rr >= 0) ? 1 : 0;
            int rid = mk ? rr : 0;
            ridS[w][m] = rid;
            mskS[w][m] = mk;
            v2f im = ((const v2f*)Imodel)[rid];
            fip.feats[w][m][0] = im.x;     // loc
            fip.feats[w][m][1] = im.y;     // q_scale
            fip.feats[w][m][2] = I[nr];
            fip.feats[w][m][3] = SigI[nr];
#pragma unroll
            for (int j = 0; j < 4; ++j) fip.feats[w][m][4 + j] = metadata[nr * 8 + j];
        } else {
#pragma unroll
            for (int j = 4; j < 8; ++j) fip.feats[w][m][4 + j] = metadata[nr * 8 + j];
            fip.feats[w][m][12] = 0.f;     // d_inv (faithful: zeros)
            fip.feats[w][m][13] = wav_exp[o];
            fip.feats[w][m][14] = 0.f;     // K pad
            fip.feats[w][m][15] = 0.f;
        }
    }
    __syncthreads();

    // GEMM1: feats[16x16] x W1[16x128] -> relu -> hidS
#pragma unroll
    for (int nt = 0; nt < 8; ++nt) {
        v8f acc = {};
#pragma unroll
        for (int k0 = 0; k0 < 16; k0 += 4) {
            v2f a = *(const v2f*)&fip.feats[w][li][k0 + 2 * half];
            v2f b = W1pS[(k0 / 2 + half) * DHID + nt * 16 + li];
            acc = wmma4(a, b, acc);
        }
        const float bv = b1[nt * 16 + li];
#pragma unroll
        for (int r = 0; r < 8; ++r) {
            float h = acc[r] + bv;
            hidS[w][r + 8 * half][nt * 16 + li] = (h > 0.f) ? h : 0.f;
        }
    }
    __syncthreads();

    // GEMM2: hid[16x128] x W2[128x32] -> scale; fuse z-gather, mask, harmonic pair-sum
#pragma unroll
    for (int tt = 0; tt < 2; ++tt) {
        v8f acc = {};
#pragma unroll 8
        for (int k0 = 0; k0 < DHID; k0 += 4) {
            v2f a = *(const v2f*)&hidS[w][li][k0 + 2 * half];
            v2f b = W2pS[(k0 / 2 + half) * MC + tt * 16 + li];
            acc = wmma4(a, b, acc);
        }
        const float bv = b2[tt * 16 + li];
        const int mc = tt * 16 + li;
        float prod[8];
#pragma unroll
        for (int r = 0; r < 8; ++r) {
            const int m  = r + 8 * half;
            const float sc = acc[r] + bv;
            prod[r] = mskS[w][m] ? zT[(size_t)ridS[w][m] * MC + mc] * sc : 0.f;
        }
#pragma unroll
        for (int g = 0; g < 4; ++g)
            fip.ipred[w][g + 4 * half][mc] = prod[2 * g] + prod[2 * g + 1];
    }
    __syncthreads();

    // Per-(op,n,r) group epilogue: mean over MC of logp and Ipred
    if (l < 8) {
        const int gg  = l;
        const int grp = (base >> 1) + gg;
        const int nr  = grp & (NR - 1);
        const float Iv = I[nr], Sv = SigI[nr];
        const int om = mskS[w][2 * gg] | mskS[w][2 * gg + 1];
        float slp = 0.f, sip = 0.f;
#pragma unroll 8
        for (int mc = 0; mc < MC; ++mc) {
            float ip = fip.ipred[w][gg][mc];
            sip += ip;
            float rz = (ip - Iv) / Sv;
            slp += rz * rz;
        }
        ws_ll[grp]    = om ? (-0.5f * slp * (1.f / MC) - logf(Sv) - 0.5f * LOG2PI) : 0.f;
        ws_ipm[grp]   = sip * (1.f / MC);
        ws_maskf[grp] = om ? 1.f : 0.f;
    }
}

// ---------------- Kernel 3: per-(op,image) ll_img = sum(ll)/sum(obs_mask)
__global__ __launch_bounds__(256) void k_llimg(const float* __restrict__ ws_ll,
                                               const float* __restrict__ ws_maskf,
                                               float* __restrict__ llimg) {
    __shared__ float sm[256];
    const int b = blockIdx.x;  // op*8 + n
    float sll = 0.f, scnt = 0.f;
    for (int r = threadIdx.x; r < R_REFL; r += 256) {
        const int g = b * R_REFL + r;
        sll  += ws_ll[g];
        scnt += ws_maskf[g];
    }
    sm[threadIdx.x] = sll; __syncthreads();
    for (int s = 128; s > 0; s >>= 1) { if (threadIdx.x < s) sm[threadIdx.x] += sm[threadIdx.x + s]; __syncthreads(); }
    const float tll = sm[0]; __syncthreads();
    sm[threadIdx.x] = scnt; __syncthreads();
    for (int s = 128; s > 0; s >>= 1) { if (threadIdx.x < s) sm[threadIdx.x] += sm[threadIdx.x + s]; __syncthreads(); }
    if (threadIdx.x == 0) llimg[b] = tll / sm[0];
}

// ---------------- Kernel 4: KL, argmax over ops, elbo, op_idx
__global__ __launch_bounds__(256) void k_final(const float* __restrict__ klp,
                                               const float* __restrict__ llimg,
                                               float* __restrict__ outv,
                                               int* __restrict__ opidx) {
    __shared__ float sm[256];
    float v = 0.f;
    for (int i = threadIdx.x; i < 1024; i += 256) v += klp[i];
    sm[threadIdx.x] = v; __syncthreads();
    for (int s = 128; s > 0; s >>= 1) { if (threadIdx.x < s) sm[threadIdx.x] += sm[threadIdx.x + s]; __syncthreads(); }
    if (threadIdx.x == 0) {
        const float kl = sm[0] / (float)NREFL;
        float sb = 0.f;
        for (int n = 0; n < N_IMG; ++n) {
            const float l0 = llimg[n], l1 = llimg[N_IMG + n];
            const int oi = (l1 > l0) ? 1 : 0;   // argmax, first-max on tie
            sb += oi ? l1 : l0;
            opidx[n] = oi;
            outv[3 + n] = (float)oi;
        }
        outv[0] = -(sb / (float)N_IMG) + kl;
        outv[1] = kl;
    }
}

// ---------------- Kernel 5: weighted-Pearson moment partials (per-block)
__global__ __launch_bounds__(256) void k_ccpart(const float* __restrict__ I,
                                                const float* __restrict__ SigI,
                                                const float* __restrict__ ipm,
                                                const float* __restrict__ maskf,
                                                const int* __restrict__ opidx,
                                                float* __restrict__ ccp) {
    __shared__ float sm[256];
    float a0 = 0, a1 = 0, a2 = 0, a3 = 0, a4 = 0, a5 = 0;
    for (int i = blockIdx.x * 256 + threadIdx.x; i < NR; i += 256 * 256) {
        const int n = i >> 14;  // R = 16384
        const float y = ipm[opidx[n] * NR + i];
        const float x = I[i];
        const float s = SigI[i];
        const float wg = (maskf[(OPS_N - 1) * NR + i] != 0.f) ? 1.f / (s * s) : 0.f;
        a0 += wg; a1 += wg * x; a2 += wg * y;
        a3 += wg * x * x; a4 += wg * y * y; a5 += wg * x * y;
    }
    float vals[6] = {a0, a1, a2, a3, a4, a5};
    for (int j = 0; j < 6; ++j) {
        sm[threadIdx.x] = vals[j]; __syncthreads();
        for (int s = 128; s > 0; s >>= 1) { if (threadIdx.x < s) sm[threadIdx.x] += sm[threadIdx.x + s]; __syncthreads(); }
        if (threadIdx.x == 0) ccp[blockIdx.x * 6 + j] = sm[0];
        __syncthreads();
    }
}

// ---------------- Kernel 6: finalize cc
__global__ __launch_bounds__(256) void k_ccfin(const float* __restrict__ ccp,
                                               float* __restrict__ outv) {
    __shared__ float sm[256];
    float tot[6];
    for (int j = 0; j < 6; ++j) {
        sm[threadIdx.x] = ccp[threadIdx.x * 6 + j]; __syncthreads();
        for (int s = 128; s > 0; s >>= 1) { if (threadIdx.x < s) sm[threadIdx.x] += sm[threadIdx.x + s]; __syncthreads(); }
        tot[j] = sm[0]; __syncthreads();
    }
    if (threadIdx.x == 0) {
        const float z  = 1.f / tot[0];
        const float mx = z * tot[1], my = z * tot[2];
        const float cxy = z * tot[5] - mx * my;
        const float cx  = z * tot[3] - mx * mx;
        const float cy  = z * tot[4] - my * my;
        outv[2] = cxy / sqrtf(cx * cy);
    }
}

extern "C" void kernel_launch(void* const* d_in, const int* in_sizes, int n_in,
                              void* d_out, int out_size, void* d_ws, size_t ws_size,
                              hipStream_t stream) {
    (void)in_sizes; (void)n_in; (void)out_size; (void)ws_size;
    // setup_inputs order: asu_id, hkl, I, SigI, metadata, wavelength, refl_id,
    //                     dHKL, wav_exp, loc, raw_scale, W1, b1, W2, b2, eps
    const float* I         = (const float*)d_in[2];
    const float* SigI      = (const float*)d_in[3];
    const float* metadata  = (const float*)d_in[4];
    const int*   refl_id   = (const int*)d_in[6];
    const float* wav_exp   = (const float*)d_in[8];
    const float* loc       = (const float*)d_in[9];
    const float* raw_scale = (const float*)d_in[10];
    const float* W1        = (const float*)d_in[11];
    const float* b1        = (const float*)d_in[12];
    const float* W2        = (const float*)d_in[13];
    const float* b2        = (const float*)d_in[14];
    const float* eps       = (const float*)d_in[15];

    float* ws       = (float*)d_ws;
    float* zT       = ws;                                   // NREFL*MC   (33.5 MB)
    float* Imodel   = zT + (size_t)NREFL * MC;              // NREFL*2
    float* klp      = Imodel + (size_t)NREFL * 2;           // 1024
    float* ws_ll    = klp + 1024;                           // NGRP
    float* ws_ipm   = ws_ll + NGRP;                         // NGRP
    float* ws_maskf = ws_ipm + NGRP;                        // NGRP
    float* llimg    = ws_maskf + NGRP;                      // 16
    float* ccp      = llimg + 16;                           // 256*6
    int*   opidx    = (int*)(ccp + 256 * 6);                // 8

    float* outv = (float*)d_out;

    k_sample<<<NREFL / 256, 256, 0, stream>>>(loc, raw_scale, eps, zT, Imodel, klp);
    k_main<<<NOBS / 64, 128, 0, stream>>>(I, SigI, metadata, refl_id, wav_exp,
                                          W1, b1, W2, b2, zT, Imodel,
                                          ws_ll, ws_ipm, ws_maskf);
    k_llimg<<<16, 256, 0, stream>>>(ws_ll, ws_maskf, llimg);
    k_final<<<1, 256, 0, stream>>>(klp, llimg, outv, opidx);
    k_ccpart<<<256, 256, 0, stream>>>(I, SigI, ws_ipm, ws_maskf, opidx, ccp);
    k_ccfin<<<1, 256, 0, stream>>>(ccp, outv);
}